// SpatialModulation_42984032698742
// MI455X (gfx1250) — compile-verified
//
#include <hip/hip_runtime.h>

typedef float f32x4 __attribute__((ext_vector_type(4)));

#define B_ 16
#define C_ 256
#define H_ 128
#define W_ 128
#define TILE_ROWS 4                   // interior rows per block
#define HALO_ROWS 6                   // interior + 1 row above/below
#define CH_PER 4                      // channels per async pipeline stage
#define STAGES (C_ / CH_PER)          // 64
#define POS (HALO_ROWS * W_)          // 768 halo positions per channel
#define THREADS 256
#define NBUF 3                        // triple-buffered async pipeline

// Low 32 bits of a generic pointer into LDS == wave-relative LDS byte address
__device__ __forceinline__ unsigned lds_addr_of(const void* p) {
    return (unsigned)(unsigned long long)p;
}

__global__ __launch_bounds__(THREADS)
void spatial_mod_fused(const float* __restrict__ x,
                       const float* __restrict__ w,
                       float* __restrict__ out) {
    __shared__ float buf[NBUF][CH_PER * POS]; // 3 x 12 KB async staging buffers
    __shared__ float y_avg[POS];              // halo avg map
    __shared__ float y_max[POS];              // halo max map
    __shared__ float msig[TILE_ROWS * W_];    // 1 + sigmoid(y3) for interior

    const int tid  = threadIdx.x;
    const int tile = blockIdx.x;              // 0..31 -> rows [tile*4, tile*4+3]
    const int b    = blockIdx.y;              // 0..15
    const int h0   = tile * TILE_ROWS;

    const float* xb = x   + (size_t)b * C_ * H_ * W_;
    float*       ob = out + (size_t)b * C_ * H_ * W_;

    // Issue one pipeline stage: 4 channels x 6 halo rows x 128 cols,
    // as 768 b128 async copies (3 per thread). Rows outside the image are
    // clamped (fetch valid data from a safe address); their y is forced to 0.
    auto issue_stage = [&](int c0, float* dst) {
#pragma unroll
        for (int j = 0; j < 3; ++j) {
            int q    = tid + j * THREADS;     // 0..767 float4-chunk id
            int ch   = q / 192;               // 0..3 channel within stage
            int rp   = q - ch * 192;          // 0..191
            int row  = rp >> 5;               // 0..5 halo row
            int col4 = rp & 31;               // 0..31 float4 column
            int gr   = h0 - 1 + row;
            gr = gr < 0 ? 0 : (gr > H_ - 1 ? H_ - 1 : gr);
            const float* gp = xb + (((size_t)(c0 + ch) * H_ + gr) << 7) + (col4 << 2);
            unsigned la = lds_addr_of(dst + ch * POS + row * W_ + (col4 << 2));
            asm volatile("global_load_async_to_lds_b128 %0, %1, off"
                         :: "v"(la), "v"(gp) : "memory");
        }
    };

    float sum[3] = {0.f, 0.f, 0.f};
    float mx[3];
    mx[0] = mx[1] = mx[2] = -3.402823466e+38f;

    // Prologue: prefetch stages 0 and 1 (prefetch distance = 2).
    issue_stage(0 * CH_PER, &buf[0][0]);
    issue_stage(1 * CH_PER, &buf[1][0]);

    int cur = 0;                              // buffer holding stage s
    for (int s = 0; s < STAGES; ++s) {
        // Only stages s and s+1 can be outstanding (3 async ops each, retired
        // in order) -> asynccnt <= 3 proves stage s has landed in LDS.
        if (s < STAGES - 1) {
            asm volatile("s_wait_asynccnt 0x3" ::: "memory");
        } else {
            asm volatile("s_wait_asynccnt 0x0" ::: "memory");
        }
        __syncthreads();  // all waves' stage-s data visible; all waves have
                          // finished reading buf[(s-1)%3] == issue target below
        if (s + 2 < STAGES) {
            int iss = (cur == 0) ? 2 : cur - 1;   // (s+2) % 3
            issue_stage((s + 2) * CH_PER, &buf[iss][0]);
        }
        const float* cb = &buf[cur][0];
#pragma unroll
        for (int ch = 0; ch < CH_PER; ++ch) {
#pragma unroll
            for (int k = 0; k < 3; ++k) {
                float v = cb[ch * POS + tid + k * THREADS];
                sum[k] += v;
                mx[k]   = fmaxf(mx[k], v);
            }
        }
        cur = (cur == NBUF - 1) ? 0 : cur + 1;
    }
    __syncthreads();  // reduction registers complete before y maps are read

    // y halo maps; zero-pad rows outside the image (matches jnp.pad on y)
#pragma unroll
    for (int k = 0; k < 3; ++k) {
        int pos = tid + k * THREADS;
        int row = pos >> 7;
        int gr  = h0 - 1 + row;
        bool valid = (gr >= 0) && (gr < H_);
        y_avg[pos] = valid ? sum[k] * (1.0f / C_) : 0.0f;
        y_max[pos] = valid ? mx[k] : 0.0f;
    }
    __syncthreads();

    const float* wb = w + b * 8;
    float w0 = wb[0], w1 = wb[1], w2 = wb[2], w3 = wb[3];
    float w4 = wb[4], w5 = wb[5], w6 = wb[6], w7 = wb[7];
    float cen0 = -(w0 + w1 + w2 + w3);
    float cen1 = -(w4 + w5 + w6 + w7);

    // 5-point stencil over both channels, summed; zero-pad columns
#pragma unroll
    for (int k = 0; k < 2; ++k) {
        int pos = tid + k * THREADS;          // 0..511 interior position
        int ir  = pos >> 7;                   // 0..3
        int c   = pos & 127;
        int r   = ir + 1;                     // halo row index
        float t0 = y_avg[(r - 1) * W_ + c], bo0 = y_avg[(r + 1) * W_ + c];
        float l0 = c > 0      ? y_avg[r * W_ + c - 1] : 0.f;
        float r0 = c < W_ - 1 ? y_avg[r * W_ + c + 1] : 0.f;
        float z0 = y_avg[r * W_ + c];
        float t1 = y_max[(r - 1) * W_ + c], bo1 = y_max[(r + 1) * W_ + c];
        float l1 = c > 0      ? y_max[r * W_ + c - 1] : 0.f;
        float r1 = c < W_ - 1 ? y_max[r * W_ + c + 1] : 0.f;
        float z1 = y_max[r * W_ + c];
        float y3 = w0 * t0 + w1 * bo0 + w2 * l0 + w3 * r0 + cen0 * z0
                 + w4 * t1 + w5 * bo1 + w6 * l1 + w7 * r1 + cen1 * z1;
        msig[pos] = 1.0f + 1.0f / (1.0f + __expf(-y3));
    }
    __syncthreads();

    // Streaming pass: re-read interior x (L2-hot: this block just streamed
    // it), scale, and NT-store the output so it doesn't evict L2-resident x.
    // Each thread owns a fixed (row, col4) slot and walks 128 channels.
    {
        int qf   = tid & 127;
        int fr   = qf >> 5;                   // interior row 0..3
        int fc4  = qf & 31;                   // float4 column
        f32x4 mv = *(const f32x4*)&msig[fr * W_ + (fc4 << 2)];
        const f32x4* x4 = (const f32x4*)xb;
        f32x4*       o4 = (f32x4*)ob;
        int cpar = tid >> 7;                  // channel parity 0/1
#pragma unroll 8
        for (int k = 0; k < 128; ++k) {
            int c   = cpar + 2 * k;
            int idx = (c * H_ + h0 + fr) * (W_ / 4) + fc4;
            f32x4 xv = x4[idx];
            f32x4 ov = xv * mv;               // x*sigmoid(y3) + x == x*(1+sig)
            __builtin_nontemporal_store(ov, &o4[idx]);
        }
    }
}

extern "C" void kernel_launch(void* const* d_in, const int* in_sizes, int n_in,
                              void* d_out, int out_size, void* d_ws, size_t ws_size,
                              hipStream_t stream) {
    const float* x = (const float*)d_in[0];              // (16,256,128,128) f32
    const float* w = (const float*)d_in[1];              // (16,8) f32
    float* out = (float*)d_out;                          // (16,256,128,128) f32
    (void)in_sizes; (void)n_in; (void)out_size; (void)d_ws; (void)ws_size;
    dim3 grid(H_ / TILE_ROWS, B_);                       // 32 x 16 = 512 blocks
    spatial_mod_fused<<<grid, THREADS, 0, stream>>>(x, w, out);
}